// LIFNeuronLayer_6760278524127
// MI455X (gfx1250) — compile-verified
//
#include <hip/hip_runtime.h>
#include <cstdint>

#define NN 256      // neurons
#define SS 64       // sensory inputs
#define BB 1024     // batch
#define STEPS 12    // ODE unfolds
#define BT 4        // batches per block
#define EPSF 1e-8f

__device__ __forceinline__ float softplus_f(float x) {
    return (x > 20.0f) ? x : log1pf(__expf(x));
}

__device__ __forceinline__ float tanh_hw(float x) {
#if __has_builtin(__builtin_amdgcn_tanhf)
    return __builtin_amdgcn_tanhf(x);
#elif __has_builtin(__builtin_amdgcn_tanh_f32)
    return __builtin_amdgcn_tanh_f32(x);
#else
    float r;
    asm volatile("v_tanh_f32 %0, %1" : "=v"(r) : "v"(x));   // gfx1250 trans op
    return r;
#endif
}

// Input t = 0.5 * x ; returns sigmoid(x) = 0.5*tanh(x/2)+0.5  (1 trans op)
__device__ __forceinline__ float sigmoid_half(float t) {
    return fmaf(0.5f, tanh_hw(t), 0.5f);
}

// One-shot parameter fusion:
//   PK[i*N+n] = { A, A*erev, 0.5*sigma, 0.5*sigma*mu }   with A = softplus(w)*mask
//   PS[s*N+n] = same for sensory params
//   vecs      = { cm_t[N], gleak_sp[N], gleak_sp*vleak[N] }
__global__ __launch_bounds__(256) void ltc_fuse(
    const float* __restrict__ gleak, const float* __restrict__ vleak,
    const float* __restrict__ cm,
    const float* __restrict__ w,    const float* __restrict__ sigma,
    const float* __restrict__ mu,   const float* __restrict__ erev,
    const float* __restrict__ sw,   const float* __restrict__ ssig,
    const float* __restrict__ smu,  const float* __restrict__ serev,
    const float* __restrict__ mask, const float* __restrict__ smask,
    float4* __restrict__ PK, float4* __restrict__ PS, float* __restrict__ vecs)
{
    int idx = blockIdx.x * blockDim.x + threadIdx.x;
    if (idx < NN * NN) {
        float A  = softplus_f(w[idx]) * mask[idx];
        float sg = sigma[idx];
        PK[idx] = make_float4(A, A * erev[idx], 0.5f * sg, 0.5f * sg * mu[idx]);
    }
    if (idx < SS * NN) {
        float A  = softplus_f(sw[idx]) * smask[idx];
        float sg = ssig[idx];
        PS[idx] = make_float4(A, A * serev[idx], 0.5f * sg, 0.5f * sg * smu[idx]);
    }
    if (idx < NN) {
        float g = softplus_f(gleak[idx]);
        vecs[idx]          = softplus_f(cm[idx]) * (float)STEPS;  // cm_t = softplus(cm)/(1/12)
        vecs[NN + idx]     = g;
        vecs[2 * NN + idx] = g * vleak[idx];
    }
}

__global__ __launch_bounds__(NN) void ltc_main(
    const float4* __restrict__ PK, const float4* __restrict__ PS,
    const float*  __restrict__ vecs,
    const float*  __restrict__ inputs, const float* __restrict__ state,
    float* __restrict__ out)
{
    // v tile stored transposed: vsh[i][b] so per-i reads are one broadcast ds_load_b128
    __shared__ __align__(16) float vsh[NN][BT];
    __shared__ __align__(16) float insh[SS][BT];

    const int n  = threadIdx.x;
    const int b0 = blockIdx.x * BT;

    // warm the packed-parameter stream (emits global_prefetch_b8)
    __builtin_prefetch(&PK[n], 0, 0);

    const float cmt = vecs[n];
    const float gl  = vecs[NN + n];
    const float glv = vecs[2 * NN + n];

    // stage inputs tile (BT*SS == 256 elements, one per thread, coalesced)
    {
        int b = threadIdx.x >> 6;   // /64
        int s = threadIdx.x & 63;
        insh[s][b] = inputs[(size_t)(b0 + b) * SS + s];
    }
    // stage state tile via async global->LDS copies (ASYNCcnt path)
#pragma unroll
    for (int b = 0; b < BT; ++b) {
        const float* gp = state + (size_t)(b0 + b) * NN + n;
        uint32_t lds_addr = (uint32_t)(uintptr_t)(&vsh[n][b]);
        asm volatile("global_load_async_to_lds_b32 %0, %1, off"
                     :: "v"(lds_addr), "v"(gp) : "memory");
    }
    asm volatile("s_wait_asynccnt 0" ::: "memory");
    __syncthreads();

    // ---- sensory accumulation (step-invariant) ----
    float an0[BT], ad0[BT];
#pragma unroll
    for (int b = 0; b < BT; ++b) { an0[b] = 0.0f; ad0[b] = 0.0f; }
    for (int s = 0; s < SS; ++s) {
        float4 p  = PS[s * NN + n];
        float4 xv = *(const float4*)(&insh[s][0]);   // LDS broadcast
        float xs[BT] = {xv.x, xv.y, xv.z, xv.w};
#pragma unroll
        for (int b = 0; b < BT; ++b) {
            float t  = fmaf(p.z, xs[b], -p.w);
            float sg = sigmoid_half(t);
            an0[b] = fmaf(p.y, sg, an0[b]);
            ad0[b] = fmaf(p.x, sg, ad0[b]);
        }
    }

    // ---- 12 ODE unfolds ----
    float vnew[BT];
    for (int step = 0; step < STEPS; ++step) {
        float an[BT], ad[BT];
#pragma unroll
        for (int b = 0; b < BT; ++b) { an[b] = an0[b]; ad[b] = ad0[b]; }

#pragma unroll 4
        for (int i = 0; i < NN; ++i) {
            float4 p  = PK[i * NN + n];                 // coalesced 512B/wave
            float4 vv = *(const float4*)(&vsh[i][0]);   // uniform ds_load_b128
            float vs[BT] = {vv.x, vv.y, vv.z, vv.w};
#pragma unroll
            for (int b = 0; b < BT; ++b) {
                float t  = fmaf(p.z, vs[b], -p.w);      // 0.5*sigma*v - 0.5*sigma*mu
                float sg = sigmoid_half(t);             // 1x v_tanh_f32 + 2 FMA
                an[b] = fmaf(p.y, sg, an[b]);           // += A*erev*sig
                ad[b] = fmaf(p.x, sg, ad[b]);           // += A*sig
            }
        }

        float4 vold4 = *(const float4*)(&vsh[n][0]);
        float vold[BT] = {vold4.x, vold4.y, vold4.z, vold4.w};
#pragma unroll
        for (int b = 0; b < BT; ++b) {
            float num = fmaf(cmt, vold[b], glv) + an[b];
            float den = cmt + gl + ad[b] + EPSF;
            vnew[b] = num * __builtin_amdgcn_rcpf(den);
        }
        __syncthreads();   // all reads of vsh complete before overwrite
        *(float4*)(&vsh[n][0]) = make_float4(vnew[0], vnew[1], vnew[2], vnew[3]);
        __syncthreads();
    }

#pragma unroll
    for (int b = 0; b < BT; ++b)
        out[(size_t)(b0 + b) * NN + n] = vnew[b];
}

extern "C" void kernel_launch(void* const* d_in, const int* in_sizes, int n_in,
                              void* d_out, int out_size, void* d_ws, size_t ws_size,
                              hipStream_t stream)
{
    const float* inputs = (const float*)d_in[0];
    const float* state  = (const float*)d_in[1];
    const float* gleak  = (const float*)d_in[2];
    const float* vleak  = (const float*)d_in[3];
    const float* cm     = (const float*)d_in[4];
    const float* w      = (const float*)d_in[5];
    const float* sigma  = (const float*)d_in[6];
    const float* mu     = (const float*)d_in[7];
    const float* erev   = (const float*)d_in[8];
    const float* sw     = (const float*)d_in[9];
    const float* ssig   = (const float*)d_in[10];
    const float* smu    = (const float*)d_in[11];
    const float* serev  = (const float*)d_in[12];
    const float* mask   = (const float*)d_in[13];
    const float* smask  = (const float*)d_in[14];

    char* ws = (char*)d_ws;
    float4* PK   = (float4*)ws;                                          // 1 MB
    float4* PS   = (float4*)(ws + (size_t)NN * NN * sizeof(float4));     // 256 KB
    float*  vecs = (float*)(ws + (size_t)NN * NN * sizeof(float4)
                               + (size_t)SS * NN * sizeof(float4));      // 3 KB

    ltc_fuse<<<(NN * NN + 255) / 256, 256, 0, stream>>>(
        gleak, vleak, cm, w, sigma, mu, erev,
        sw, ssig, smu, serev, mask, smask, PK, PS, vecs);

    ltc_main<<<BB / BT, NN, 0, stream>>>(PK, PS, vecs, inputs, state, (float*)d_out);
}